// Temporal_Attention_layer_7584912245218
// MI455X (gfx1250) — compile-verified
//
#include <hip/hip_runtime.h>
#include <math.h>

#define B_ 8
#define N_ 4096
#define C_ 8
#define T_ 256
#define NCH 32
#define NPC (N_ / NCH) /* 128 */

typedef __attribute__((ext_vector_type(2))) float v2f;
typedef __attribute__((ext_vector_type(8))) float v8f;

// workspace layout (float offsets)
#define P1SZ (B_ * NCH * C_ * T_)          /* 524288 */
#define OFF_PART1 0
#define OFF_PARTM (P1SZ)
#define OFF_XU1 (2 * P1SZ)                 /* B*C*T = 16384 */
#define OFF_M   (2 * P1SZ + B_ * C_ * T_)
#define OFF_E   (2 * P1SZ + 2 * B_ * C_ * T_) /* B*T*T = 524288 */

// ---------------- Kernel 1: single streaming pass over x (256 MB) ----------
// block = (b, n-chunk); thread = t. Computes partial xu1[b,c,t] and M[b,c,t].
__global__ __launch_bounds__(256) void k1_reduce_x(
    const float* __restrict__ x, const float* __restrict__ u1,
    const float* __restrict__ u2, const float* __restrict__ u3,
    float* __restrict__ ws) {
  __shared__ float su1[NPC];
  __shared__ float su2[C_][NPC];
  __shared__ float su3[C_];
  const int ch = blockIdx.x, b = blockIdx.y;
  const int tid = threadIdx.x; // t
  const int n0 = ch * NPC;

  if (tid < NPC) su1[tid] = u1[n0 + tid];
  for (int i = tid; i < C_ * NPC; i += 256) {
    int c = i / NPC, nl = i % NPC;
    su2[c][nl] = u2[c * N_ + n0 + nl];
  }
  if (tid < C_) su3[tid] = u3[tid];
  __syncthreads();

  float acc1[C_], accM[C_];
#pragma unroll
  for (int c = 0; c < C_; ++c) { acc1[c] = 0.f; accM[c] = 0.f; }

  const float* xb = x + ((size_t)b * N_ + n0) * (size_t)(C_ * T_) + tid;
  for (int nl = 0; nl < NPC; ++nl) {
    const float* xp = xb + (size_t)nl * (C_ * T_);
    if (nl + 2 < NPC) // uniform branch; speculative prefetch 2 slabs ahead
      __builtin_prefetch(xp + 2 * (C_ * T_), 0, 0);
    float xv[C_];
#pragma unroll
    for (int c = 0; c < C_; ++c) xv[c] = xp[c * T_];
    const float u1n = su1[nl];
    float rhsv = 0.f;
#pragma unroll
    for (int c = 0; c < C_; ++c) {
      acc1[c] = fmaf(u1n, xv[c], acc1[c]);       // partial xu1[b, t, c]
      rhsv = fmaf(xv[c], su3[c], rhsv);          // rhs[b, n, t]
    }
#pragma unroll
    for (int c = 0; c < C_; ++c)
      accM[c] = fmaf(su2[c][nl], rhsv, accM[c]); // partial M[b, c, t]
  }
  float* p1 = ws + OFF_PART1 + (size_t)(b * NCH + ch) * (C_ * T_) + tid;
  float* pM = ws + OFF_PARTM + (size_t)(b * NCH + ch) * (C_ * T_) + tid;
#pragma unroll
  for (int c = 0; c < C_; ++c) { p1[c * T_] = acc1[c]; pM[c * T_] = accM[c]; }
}

// ---------------- Kernel 2: reduce chunk partials ---------------------------
__global__ __launch_bounds__(256) void k2_reduce_parts(float* __restrict__ ws) {
  const int BCT = B_ * C_ * T_;
  int idx = blockIdx.x * 256 + threadIdx.x; // < 2*BCT
  int which = idx >= BCT;
  int r = which ? idx - BCT : idx;
  int b = r / (C_ * T_);
  int rem = r % (C_ * T_);
  const float* src = ws + (which ? OFF_PARTM : OFF_PART1);
  float s = 0.f;
  for (int chn = 0; chn < NCH; ++chn)
    s += src[(size_t)(b * NCH + chn) * (C_ * T_) + rem];
  ws[(which ? OFF_M : OFF_XU1) + r] = s;
}

// ---------------- Kernel 3: P = sigmoid(xu1^T M + bias); E = v @ P (WMMA) ---
// grid = (j-block 0..3, batch 0..7); 512 threads = 16 waves (wave32).
// P stored k-pair interleaved so each WMMA B operand {P[kk][j], P[kk+1][j]}
// is one aligned contiguous ds_load_b64 into an even VGPR pair.
__global__ __launch_bounds__(512) void k3_sig_gemm(
    const float* __restrict__ bias, const float* __restrict__ v,
    float* __restrict__ ws) {
  __shared__ float lP[T_ / 2][64][2]; // 64 KB: [k/2][j'][k&1]
  const int jb = blockIdx.x;
  const int b  = blockIdx.y;
  const int tid = threadIdx.x;
  const int jbase = jb * 64;
  const float* xu1 = ws + OFF_XU1 + (size_t)b * (C_ * T_); // [c][t]
  const float* Mm  = ws + OFF_M   + (size_t)b * (C_ * T_); // [c][s]

  // Phase 1: fill P tile (unique work, 32 entries/thread)
  for (int e = tid; e < T_ * 64; e += 512) {
    int k = e >> 6;
    int jl = e & 63;
    int j = jbase + jl;
    float p = bias[k * T_ + j];
#pragma unroll
    for (int c = 0; c < C_; ++c)
      p = fmaf(xu1[c * T_ + k], Mm[c * T_ + j], p);
    lP[k >> 1][jl][k & 1] = 1.0f / (1.0f + __expf(-p));
  }
  __syncthreads();

  // Phase 2: E[i0..i0+15][jbase..jbase+63] via V_WMMA_F32_16X16X4_F32.
  // A 16x4 layout: VGPR0 = K0 (lanes 0-15) / K2 (lanes 16-31); VGPR1 = K1/K3.
  const int wave = tid >> 5;
  const int lane = tid & 31;
  const int half = lane >> 4; // 0/1, branch-free so EXEC stays all-1s
  const int m = lane & 15;
  const int i0 = wave * 16;

  v8f acc0 = {}, acc1 = {}, acc2 = {}, acc3 = {};
  for (int k0 = 0; k0 < T_; k0 += 4) {
    const int kk = k0 + half * 2;        // even -> 8B-aligned v offset
    const int kp = (k0 >> 1) + half;     // interleaved pair row
    v2f a  = *(const v2f*)(v + (i0 + m) * T_ + kk); // global_load_b64
    v2f b0 = *(const v2f*)(&lP[kp][ 0 + m][0]);     // ds_load_b64
    v2f b1 = *(const v2f*)(&lP[kp][16 + m][0]);
    v2f b2 = *(const v2f*)(&lP[kp][32 + m][0]);
    v2f b3 = *(const v2f*)(&lP[kp][48 + m][0]);
    acc0 = __builtin_amdgcn_wmma_f32_16x16x4_f32(false, a, false, b0, (short)0, acc0, false, false);
    acc1 = __builtin_amdgcn_wmma_f32_16x16x4_f32(false, a, false, b1, (short)0, acc1, false, false);
    acc2 = __builtin_amdgcn_wmma_f32_16x16x4_f32(false, a, false, b2, (short)0, acc2, false, false);
    acc3 = __builtin_amdgcn_wmma_f32_16x16x4_f32(false, a, false, b3, (short)0, acc3, false, false);
  }

  // Phase 3: spill E tiles. C/D layout: VGPR p -> M=p (lanes 0-15) / M=p+8.
  float* E = ws + OFF_E + (size_t)b * (T_ * T_);
  const int rowBase = i0 + half * 8;
  const int colBase = jbase + m;
#pragma unroll
  for (int p = 0; p < 8; ++p) {
    E[(rowBase + p) * T_ + colBase +  0] = acc0[p];
    E[(rowBase + p) * T_ + colBase + 16] = acc1[p];
    E[(rowBase + p) * T_ + colBase + 32] = acc2[p];
    E[(rowBase + p) * T_ + colBase + 48] = acc3[p];
  }
}

// ---------------- Kernel 4: row softmax (one wave32 per row) ----------------
__global__ __launch_bounds__(256) void k4_softmax(const float* __restrict__ E,
                                                  float* __restrict__ out) {
  const int wave = threadIdx.x >> 5;
  const int lane = threadIdx.x & 31;
  const int row = blockIdx.x * 8 + wave; // 0..B*T-1
  const float* Er = E + (size_t)row * T_;
  float vals[8];
#pragma unroll
  for (int q = 0; q < 8; ++q) vals[q] = Er[lane + 32 * q];
  float mx = vals[0];
#pragma unroll
  for (int q = 1; q < 8; ++q) mx = fmaxf(mx, vals[q]);
#pragma unroll
  for (int off = 16; off >= 1; off >>= 1)
    mx = fmaxf(mx, __shfl_xor(mx, off, 32));
  float sum = 0.f;
#pragma unroll
  for (int q = 0; q < 8; ++q) { vals[q] = __expf(vals[q] - mx); sum += vals[q]; }
#pragma unroll
  for (int off = 16; off >= 1; off >>= 1)
    sum += __shfl_xor(sum, off, 32);
  const float inv = 1.0f / sum;
  float* Or = out + (size_t)row * T_;
#pragma unroll
  for (int q = 0; q < 8; ++q) Or[lane + 32 * q] = vals[q] * inv;
}

extern "C" void kernel_launch(void* const* d_in, const int* in_sizes, int n_in,
                              void* d_out, int out_size, void* d_ws, size_t ws_size,
                              hipStream_t stream) {
  const float* x    = (const float*)d_in[0];
  const float* u1   = (const float*)d_in[1];
  const float* u2   = (const float*)d_in[2];
  const float* u3   = (const float*)d_in[3];
  const float* bias = (const float*)d_in[4];
  const float* v    = (const float*)d_in[5];
  float* ws  = (float*)d_ws;
  float* out = (float*)d_out;

  k1_reduce_x<<<dim3(NCH, B_), 256, 0, stream>>>(x, u1, u2, u3, ws);
  k2_reduce_parts<<<dim3((2 * B_ * C_ * T_) / 256), 256, 0, stream>>>(ws);
  k3_sig_gemm<<<dim3(T_ / 64, B_), 512, 0, stream>>>(bias, v, ws);
  k4_softmax<<<dim3((B_ * T_) / 8), 256, 0, stream>>>(ws + OFF_E, out);
}